// knrm_90701119356944
// MI455X (gfx1250) — compile-verified
//
#include <hip/hip_runtime.h>
#include <math.h>

#define B 64
#define LQ 20
#define LD 512
#define DIM 300
#define NB 11
#define EQ 5
#define ED 10
#define DESCL 20
#define KC 10

typedef __attribute__((ext_vector_type(2))) float v2f;
typedef __attribute__((ext_vector_type(8))) float v8f;

__constant__ float c_MU[11]  = {1.0f,0.9f,0.7f,0.5f,0.3f,0.1f,-0.1f,-0.3f,-0.5f,-0.7f,-0.9f};
__constant__ float c_IS2[11] = {500000.0f,50.0f,50.0f,50.0f,50.0f,50.0f,50.0f,50.0f,50.0f,50.0f,50.0f};

// ---------------- embedding gather: one block per row ----------------
__global__ void gather_rows_kernel(const int* __restrict__ ids, const float* __restrict__ emb,
                                   float* __restrict__ out, int dim) {
    int row = blockIdx.x;
    int id = ids[row];
    const float* src = emb + (size_t)id * dim;
    float* dst = out + (size_t)row * dim;
    for (int j = threadIdx.x; j < dim; j += blockDim.x) dst[j] = src[j];
}

// ---------------- bag-of-words projection: bow[b,o] = (sum_l emb_l) . W_o + L*b_o ----
__global__ void bow_kernel(const float* __restrict__ wbuf, const float* __restrict__ bow_W,
                           const float* __restrict__ bow_b, float* __restrict__ bow_out, int L) {
    int b = blockIdx.x;
    __shared__ float sw[DIM];
    for (int j = threadIdx.x; j < DIM; j += blockDim.x) {
        float s = 0.0f;
        const float* p = wbuf + (size_t)b * L * DIM + j;
        for (int l = 0; l < L; ++l) s += p[(size_t)l * DIM];
        sw[j] = s;
    }
    __syncthreads();
    if (threadIdx.x < 128) {
        int o = threadIdx.x;
        const float* wr = bow_W + (size_t)o * DIM;
        float acc = 0.0f;
        for (int j = 0; j < DIM; ++j) acc += sw[j] * wr[j];
        bow_out[b * 128 + o] = acc + (float)L * bow_b[o];
    }
}

// ---------------- word-conv as WMMA f32 GEMM ----------------
// A[m, j] = Abuf[item][m*300 + j], j < K = k*300  (sliding-window == strided row view)
// C[m, o] = relu(sum_j A[m,j] * W[o,j] + bias[o]); mode 0: l2-norm rows, store (nb,Lout,128)
//                                                  mode 1: max over 16 rows, store (nb,128)
__global__ void conv_wmma_kernel(const float* __restrict__ Abuf, int Astride,
                                 const float* __restrict__ W, const float* __restrict__ bias,
                                 float* __restrict__ out, int Lout, int K, int mode) {
    int item = blockIdx.y;
    int m0 = blockIdx.x * 16;
    int tid = threadIdx.x;
    int wave = tid >> 5, lane = tid & 31;
    int h = lane >> 4, r = lane & 15;

    int m = m0 + r;
    if (m >= Lout) m = Lout - 1;                 // clamp (results discarded at store)
    const float* Arow = Abuf + (size_t)item * Astride + (size_t)m * DIM;
    const float* Wrow = W + (size_t)(wave * 16 + r) * K;
    int koff = 2 * h;

    v8f c = {};
#pragma unroll 2
    for (int kk = 0; kk < K; kk += 4) {
        v2f a = *(const v2f*)(Arow + kk + koff);
        v2f w = *(const v2f*)(Wrow + kk + koff);
        c = __builtin_amdgcn_wmma_f32_16x16x4_f32(false, a, false, w, (short)0, c, false, false);
    }

    __shared__ float tile[16][136];
    float bval = bias[wave * 16 + r];
#pragma unroll
    for (int v = 0; v < 8; ++v) {
        float x = c[v] + bval;
        tile[v + 8 * h][wave * 16 + r] = x > 0.0f ? x : 0.0f;
    }
    __syncthreads();

    if (mode == 0) {
        // each wave l2-normalizes 2 rows
        for (int ml = wave * 2; ml < wave * 2 + 2; ++ml) {
            int mm = m0 + ml;
            if (mm >= Lout) continue;
            float s = 0.0f;
#pragma unroll
            for (int j = 0; j < 4; ++j) { float x = tile[ml][lane + 32 * j]; s += x * x; }
#pragma unroll
            for (int off = 16; off >= 1; off >>= 1) s += __shfl_xor(s, off, 32);
            float inv = 1.0f / fmaxf(sqrtf(s), 1e-10f);
#pragma unroll
            for (int j = 0; j < 4; ++j)
                out[((size_t)item * Lout + mm) * 128 + lane + 32 * j] = tile[ml][lane + 32 * j] * inv;
        }
    } else {
        if (tid < 128) {
            float mx = tile[0][tid];
#pragma unroll
            for (int ml = 1; ml < 16; ++ml) mx = fmaxf(mx, tile[ml][tid]);
            out[(size_t)item * 128 + tid] = mx;
        }
    }
}

// ---------------- entity branch combine: ent_emb + conv-max + attention-weighted cars, l2 ----
__global__ void entity_kernel(const int* __restrict__ ids_e, const int* __restrict__ ids_car,
                              const float* __restrict__ ent_emb, const float* __restrict__ car_emb,
                              const float* __restrict__ bow, const float* __restrict__ econv,
                              float* __restrict__ out, int n) {
    int idx = blockIdx.x;          // b*n + e
    int b = idx / n;
    int t = threadIdx.x;           // 128 threads
    __shared__ float sc[KC];
    __shared__ float red[128];
    float bw = bow[b * 128 + t];
    int cids[KC];
    for (int c = 0; c < KC; ++c) cids[c] = ids_car[(size_t)idx * KC + c];
    for (int c = 0; c < KC; ++c) {
        float p = bw * car_emb[(size_t)cids[c] * 128 + t];
        __syncthreads();
        red[t] = p;
        for (int off = 64; off >= 1; off >>= 1) { __syncthreads(); if (t < off) red[t] += red[t + off]; }
        __syncthreads();
        if (t == 0) sc[c] = red[0];
    }
    __syncthreads();
    float mx = sc[0];
    for (int c = 1; c < KC; ++c) mx = fmaxf(mx, sc[c]);
    float att[KC]; float ssum = 0.0f;
    for (int c = 0; c < KC; ++c) { att[c] = __expf(sc[c] - mx); ssum += att[c]; }
    float inv = 1.0f / ssum;
    float ew = 0.0f;
    for (int c = 0; c < KC; ++c) ew += att[c] * inv * car_emb[(size_t)cids[c] * 128 + t];
    float val = ent_emb[(size_t)ids_e[idx] * 128 + t] + econv[(size_t)idx * 128 + t] + ew;
    __syncthreads();
    red[t] = val * val;
    for (int off = 64; off >= 1; off >>= 1) { __syncthreads(); if (t < off) red[t] += red[t + off]; }
    __syncthreads();
    float nrm = fmaxf(sqrtf(red[0]), 1e-10f);
    out[(size_t)idx * 128 + t] = val / nrm;
}

// ---------------- RBF kernel pooling: one block per (pool, batch) ----------------
__global__ void pool_kernel(const float* __restrict__ qu, const float* __restrict__ qb2,
                            const float* __restrict__ qt2, const float* __restrict__ qs2,
                            const float* __restrict__ du, const float* __restrict__ db2,
                            const float* __restrict__ dt2, const float* __restrict__ ds2,
                            const float* __restrict__ qwm, const float* __restrict__ qem,
                            const float* __restrict__ dwm, const float* __restrict__ dem,
                            float* __restrict__ feats) {
    const int qsel_t[16] = {0,0,0,1,2,1,1,2,2,3,3,3,0,1,2,3};
    const int dsel_t[16] = {0,2,1,0,0,1,2,1,2,0,1,2,3,3,3,3};
    int pool = blockIdx.x, b = blockIdx.y;
    int qsel = qsel_t[pool], dsel = dsel_t[pool];
    const float* qp; int Lq_; const float* mqp;
    if (qsel == 0)      { qp = qu;  Lq_ = 20; mqp = qwm + b * LQ; }
    else if (qsel == 1) { qp = qb2; Lq_ = 19; mqp = qwm + b * LQ; }
    else if (qsel == 2) { qp = qt2; Lq_ = 18; mqp = qwm + b * LQ; }
    else                { qp = qs2; Lq_ = 5;  mqp = qem + b * EQ; }
    const float* dpp; int Ld_; const float* mdp;
    if (dsel == 0)      { dpp = du;  Ld_ = 512; mdp = dwm + b * LD; }
    else if (dsel == 1) { dpp = db2; Ld_ = 511; mdp = dwm + b * LD; }
    else if (dsel == 2) { dpp = dt2; Ld_ = 510; mdp = dwm + b * LD; }
    else                { dpp = ds2; Ld_ = 10;  mdp = dem + b * ED; }
    const float* qbase = qp + (size_t)b * Lq_ * 128;
    const float* dbase = dpp + (size_t)b * Ld_ * 128;

    int wave = threadIdx.x >> 5, lane = threadIdx.x & 31;
    float4 qreg[3]; int qposs[3]; int nq = 0;
    for (int qpos = wave; qpos < Lq_; qpos += 8) {
        qreg[nq] = *(const float4*)(qbase + (size_t)qpos * 128 + lane * 4);
        qposs[nq] = qpos; ++nq;
    }
    float mu  = (lane < NB) ? c_MU[lane]  : 0.0f;
    float is2 = (lane < NB) ? c_IS2[lane] : 0.0f;
    float acc0 = 0.0f, acc1 = 0.0f, acc2 = 0.0f;

    int ldlim = (nq > 0) ? Ld_ : 0;
    for (int dpos = 0; dpos < ldlim; ++dpos) {
        float4 dv = *(const float4*)(dbase + (size_t)dpos * 128 + lane * 4);
        float mdv = mdp[dpos];
        {
            float p = qreg[0].x*dv.x + qreg[0].y*dv.y + qreg[0].z*dv.z + qreg[0].w*dv.w;
#pragma unroll
            for (int off = 16; off >= 1; off >>= 1) p += __shfl_xor(p, off, 32);
            float d0 = p - mu; acc0 += __expf(-d0 * d0 * is2) * mdv;
        }
        if (nq > 1) {
            float p = qreg[1].x*dv.x + qreg[1].y*dv.y + qreg[1].z*dv.z + qreg[1].w*dv.w;
#pragma unroll
            for (int off = 16; off >= 1; off >>= 1) p += __shfl_xor(p, off, 32);
            float d1 = p - mu; acc1 += __expf(-d1 * d1 * is2) * mdv;
        }
        if (nq > 2) {
            float p = qreg[2].x*dv.x + qreg[2].y*dv.y + qreg[2].z*dv.z + qreg[2].w*dv.w;
#pragma unroll
            for (int off = 16; off >= 1; off >>= 1) p += __shfl_xor(p, off, 32);
            float d2 = p - mu; acc2 += __expf(-d2 * d2 * is2) * mdv;
        }
    }
    float f = 0.0f;
    if (nq > 0) f += __logf(fmaxf(acc0, 1e-10f)) * 0.01f * mqp[qposs[0]];
    if (nq > 1) f += __logf(fmaxf(acc1, 1e-10f)) * 0.01f * mqp[qposs[1]];
    if (nq > 2) f += __logf(fmaxf(acc2, 1e-10f)) * 0.01f * mqp[qposs[2]];

    __shared__ float wfeat[8][11];
    if (lane < NB) wfeat[wave][lane] = f;
    __syncthreads();
    if (threadIdx.x < NB) {
        float s = 0.0f;
        for (int w = 0; w < 8; ++w) s += wfeat[w][threadIdx.x];
        feats[(size_t)b * 176 + pool * NB + threadIdx.x] = s;
    }
}

// ---------------- final dense + tanh ----------------
__global__ void dense_kernel(const float* __restrict__ feats, const float* __restrict__ W,
                             const float* __restrict__ bias, float* __restrict__ out) {
    int b = threadIdx.x;
    if (b < B) {
        float acc = 0.0f;
        for (int k = 0; k < 176; ++k) acc += feats[b * 176 + k] * W[k];
        out[b] = tanhf(acc + bias[0]);
    }
}

extern "C" void kernel_launch(void* const* d_in, const int* in_sizes, int n_in,
                              void* d_out, int out_size, void* d_ws, size_t ws_size,
                              hipStream_t stream) {
    const int*   qwt  = (const int*)d_in[0];
    const float* qwm  = (const float*)d_in[1];
    const int*   qei  = (const int*)d_in[2];
    const int*   qet  = (const int*)d_in[3];
    const int*   qew  = (const int*)d_in[4];
    const float* qem  = (const float*)d_in[5];
    const int*   dwt  = (const int*)d_in[6];
    const float* dwm  = (const float*)d_in[7];
    const int*   dei  = (const int*)d_in[8];
    const int*   det  = (const int*)d_in[9];
    const int*   dew  = (const int*)d_in[10];
    const float* demm = (const float*)d_in[11];
    const float* wrd  = (const float*)d_in[12];
    const float* ent  = (const float*)d_in[13];
    const float* car  = (const float*)d_in[14];
    const float* bowW = (const float*)d_in[15];
    const float* bowb = (const float*)d_in[16];
    const float* cuW  = (const float*)d_in[17];
    const float* cub  = (const float*)d_in[18];
    const float* cbW  = (const float*)d_in[19];
    const float* cbb  = (const float*)d_in[20];
    const float* ctW  = (const float*)d_in[21];
    const float* ctb  = (const float*)d_in[22];
    const float* cdW  = (const float*)d_in[23];
    const float* cdb  = (const float*)d_in[24];
    const float* dfW  = (const float*)d_in[25];
    const float* dfb  = (const float*)d_in[26];
    float* out = (float*)d_out;

    float* p = (float*)d_ws;
    float* qwbuf = p; p += (size_t)B * LQ * DIM;
    float* dwbuf = p; p += (size_t)B * LD * DIM;
    float* qdbuf = p; p += (size_t)B * EQ * DESCL * DIM;
    float* ddbuf = p; p += (size_t)B * ED * DESCL * DIM;
    float* qbow  = p; p += B * 128;
    float* dbow  = p; p += B * 128;
    float* qec   = p; p += B * EQ * 128;
    float* dec   = p; p += B * ED * 128;
    float* qu    = p; p += (size_t)B * 20 * 128;
    float* qb    = p; p += (size_t)B * 19 * 128;
    float* qt    = p; p += (size_t)B * 18 * 128;
    float* qs    = p; p += (size_t)B * 5 * 128;
    float* du    = p; p += (size_t)B * 512 * 128;
    float* db    = p; p += (size_t)B * 511 * 128;
    float* dt    = p; p += (size_t)B * 510 * 128;
    float* ds    = p; p += (size_t)B * 10 * 128;
    float* feats = p; p += B * 176;

    gather_rows_kernel<<<B * LQ, 128, 0, stream>>>(qwt, wrd, qwbuf, DIM);
    gather_rows_kernel<<<B * LD, 128, 0, stream>>>(dwt, wrd, dwbuf, DIM);
    gather_rows_kernel<<<B * EQ * DESCL, 128, 0, stream>>>(qet, wrd, qdbuf, DIM);
    gather_rows_kernel<<<B * ED * DESCL, 128, 0, stream>>>(det, wrd, ddbuf, DIM);

    bow_kernel<<<B, 256, 0, stream>>>(qwbuf, bowW, bowb, qbow, LQ);
    bow_kernel<<<B, 256, 0, stream>>>(dwbuf, bowW, bowb, dbow, LD);

    conv_wmma_kernel<<<dim3(2, B), 256, 0, stream>>>(qwbuf, LQ * DIM, cuW, cub, qu, 20, 300, 0);
    conv_wmma_kernel<<<dim3(2, B), 256, 0, stream>>>(qwbuf, LQ * DIM, cbW, cbb, qb, 19, 600, 0);
    conv_wmma_kernel<<<dim3(2, B), 256, 0, stream>>>(qwbuf, LQ * DIM, ctW, ctb, qt, 18, 900, 0);
    conv_wmma_kernel<<<dim3(32, B), 256, 0, stream>>>(dwbuf, LD * DIM, cuW, cub, du, 512, 300, 0);
    conv_wmma_kernel<<<dim3(32, B), 256, 0, stream>>>(dwbuf, LD * DIM, cbW, cbb, db, 511, 600, 0);
    conv_wmma_kernel<<<dim3(32, B), 256, 0, stream>>>(dwbuf, LD * DIM, ctW, ctb, dt, 510, 900, 0);
    conv_wmma_kernel<<<dim3(1, B * EQ), 256, 0, stream>>>(qdbuf, DESCL * DIM, cdW, cdb, qec, 16, 1500, 1);
    conv_wmma_kernel<<<dim3(1, B * ED), 256, 0, stream>>>(ddbuf, DESCL * DIM, cdW, cdb, dec, 16, 1500, 1);

    entity_kernel<<<B * EQ, 128, 0, stream>>>(qei, qew, ent, car, qbow, qec, qs, EQ);
    entity_kernel<<<B * ED, 128, 0, stream>>>(dei, dew, ent, car, dbow, dec, ds, ED);

    pool_kernel<<<dim3(16, B), 256, 0, stream>>>(qu, qb, qt, qs, du, db, dt, ds,
                                                 qwm, qem, dwm, demm, feats);

    dense_kernel<<<1, 64, 0, stream>>>(feats, dfW, dfb, out);
}